// DaviesBouldinLossFunction_59957743452612
// MI455X (gfx1250) — compile-verified
//
#include <hip/hip_runtime.h>
#include <math.h>

// Problem constants (from reference): N=2,000,000 points, D=32 dims, C=1000 clusters.
#define N_PTS   2000000
#define DIM     32
#define CLS     1000
#define TILES   63                    // ceil(1000/16)
#define TILE_PAIRS (TILES * TILES)    // 3969 16x16 tiles of the C x C matrix
#define WAVES_PER_BLOCK 8             // 256 threads = 8 wave32

// Workspace layout (float offsets into d_ws). Total ~139 KB.
#define WS_CENT2 0        // [0, 32000)   : cent2 (C x D accumulator)
#define WS_S     32000    // [32000,33000): s accumulator (distances + per-cluster norm sums)
#define WS_SQ    33024    // 1000 floats  : row norms^2 of cent2
#define WS_SV    34048    // 1000 floats  : sqrt(s)/count
#define WS_BLK   35072    // per-block partial sums of the pair kernel (<= 512 floats)

typedef float v2f __attribute__((ext_vector_type(2)));
typedef float v8f __attribute__((ext_vector_type(8)));

// ---------------------------------------------------------------------------
// K1: initialize ws: cent2 <- centroids, s <- distances
// ---------------------------------------------------------------------------
__global__ void k_init(const float* __restrict__ centroids,
                       const float* __restrict__ distances,
                       float* __restrict__ ws) {
  int i = blockIdx.x * blockDim.x + threadIdx.x;
  if (i < CLS * DIM) ws[WS_CENT2 + i] = centroids[i];
  if (i < CLS)       ws[WS_S + i]     = distances[i];
}

// ---------------------------------------------------------------------------
// K2: cent2 += segment_sum(predicted / count[target]) staged through LDS.
// One wave handles 4 points per iteration; lane = (point g=lane>>3, 4-dim
// chunk sub=lane&7) -> float4 (b128) coalesced loads of predicted.
// Dynamic LDS: CLS*DIM floats = 128000 bytes per block.
// ---------------------------------------------------------------------------
__global__ void k_scatter_cent(const float* __restrict__ pred,
                               const float* __restrict__ count,
                               const int*   __restrict__ target,
                               float* __restrict__ ws) {
  extern __shared__ float lds_c[]; // CLS*DIM floats
  for (int i = threadIdx.x; i < CLS * DIM; i += blockDim.x) lds_c[i] = 0.0f;
  __syncthreads();

  const float4* pred4 = (const float4*)pred;
  const int lane = threadIdx.x & 31;
  const int g    = lane >> 3;       // point within quad
  const int sub  = lane & 7;        // 4-dim chunk within point
  const int wslot = blockIdx.x * WAVES_PER_BLOCK + (threadIdx.x >> 5);
  const int wtot  = gridDim.x * WAVES_PER_BLOCK;

  for (int q = wslot; q < N_PTS / 4; q += wtot) {
    int p = q * 4 + g;
    int t = target[p];
    float inv = 1.0f / count[t];
    float4 v = pred4[p * 8 + sub];
    float* dst = &lds_c[t * DIM + sub * 4];
    atomicAdd(dst + 0, v.x * inv);
    atomicAdd(dst + 1, v.y * inv);
    atomicAdd(dst + 2, v.z * inv);
    atomicAdd(dst + 3, v.w * inv);
  }
  __syncthreads();
  for (int i = threadIdx.x; i < CLS * DIM; i += blockDim.x)
    atomicAdd(&ws[WS_CENT2 + i], lds_c[i]);
}

// ---------------------------------------------------------------------------
// K3: s += segment_sum(||cent2[target] - predicted/count[target]||), staged
// through LDS (CLS floats dynamic). Second streaming pass over predicted.
// ---------------------------------------------------------------------------
__global__ void k_scatter_norm(const float* __restrict__ pred,
                               const float* __restrict__ count,
                               const int*   __restrict__ target,
                               float* __restrict__ ws) {
  extern __shared__ float lds_s[]; // CLS floats
  for (int i = threadIdx.x; i < CLS; i += blockDim.x) lds_s[i] = 0.0f;
  __syncthreads();

  const float4* pred4 = (const float4*)pred;
  const float4* cent4 = (const float4*)(ws + WS_CENT2);
  const int lane = threadIdx.x & 31;
  const int g    = lane >> 3;
  const int sub  = lane & 7;
  const int wslot = blockIdx.x * WAVES_PER_BLOCK + (threadIdx.x >> 5);
  const int wtot  = gridDim.x * WAVES_PER_BLOCK;

  for (int q = wslot; q < N_PTS / 4; q += wtot) {
    int p = q * 4 + g;
    int t = target[p];
    float inv = 1.0f / count[t];
    float4 v = pred4[p * 8 + sub];
    float4 c = cent4[t * 8 + sub];
    float dx = c.x - v.x * inv;
    float dy = c.y - v.y * inv;
    float dz = c.z - v.z * inv;
    float dw = c.w - v.w * inv;
    float part = dx * dx + dy * dy + dz * dz + dw * dw;
    // reduce over the 8 lanes of this point's group (xor stays in group)
    part += __shfl_xor(part, 1);
    part += __shfl_xor(part, 2);
    part += __shfl_xor(part, 4);
    if (sub == 0) atomicAdd(&lds_s[t], sqrtf(part));
  }
  __syncthreads();
  for (int i = threadIdx.x; i < CLS; i += blockDim.x)
    atomicAdd(&ws[WS_S + i], lds_s[i]);
}

// ---------------------------------------------------------------------------
// K4: sv[c] = sqrt(s[c]) / count[c];  sq[c] = ||cent2[c]||^2
// ---------------------------------------------------------------------------
__global__ void k_svsq(const float* __restrict__ count, float* __restrict__ ws) {
  int c = blockIdx.x * blockDim.x + threadIdx.x;
  if (c >= CLS) return;
  ws[WS_SV + c] = sqrtf(ws[WS_S + c]) / count[c];
  const float* row = ws + WS_CENT2 + c * DIM;
  float acc = 0.0f;
#pragma unroll
  for (int d = 0; d < DIM; ++d) acc += row[d] * row[d];
  ws[WS_SQ + c] = acc;
}

// ---------------------------------------------------------------------------
// K5: pairwise kernel. One wave owns one 16x16 tile of G = cent2 @ cent2^T,
// computed with 8 chained V_WMMA_F32_16X16X4_F32 (K = 32 = 8 x 4).
// A layout (ISA 7.12.2, 16x4 f32): lane m=l&15 holds row ti*16+m; lane-half
// h=l>>4 selects K pair {4k+2h, 4k+2h+1}. B (4x16) mirrors with N=l&15.
// Epilogue: d2 = sq[i]+sq[j]-2*G[i][j]; sum (sv[i]+sv[j])/sqrt(max(d2,0))
// over valid off-diagonal (i,j) < C. All padding is branchless (cndmask) so
// EXEC is all-1s at every WMMA; tile validity is wave-uniform.
// ---------------------------------------------------------------------------
__global__ void k_pair(float* __restrict__ ws) {
  __shared__ float wsum[WAVES_PER_BLOCK];
  const float* cent = ws + WS_CENT2;
  const float* sq   = ws + WS_SQ;
  const float* sv   = ws + WS_SV;

  const int waveId = threadIdx.x >> 5;
  const int lane   = threadIdx.x & 31;
  const int wid    = blockIdx.x * WAVES_PER_BLOCK + waveId;

  float acc = 0.0f;
  if (wid < TILE_PAIRS) {                 // wave-uniform: EXEC stays full
    const int ti = wid / TILES;
    const int tj = wid % TILES;
    const int m = lane & 15;
    const int h = lane >> 4;

    const int rowA = ti * 16 + m;
    const int rowB = tj * 16 + m;
    const float maskA = (rowA < CLS) ? 1.0f : 0.0f;
    const float maskB = (rowB < CLS) ? 1.0f : 0.0f;
    const int rA = (rowA < CLS) ? rowA : (CLS - 1);
    const int rB = (rowB < CLS) ? rowB : (CLS - 1);
    const v2f* Arow = (const v2f*)(cent + rA * DIM);
    const v2f* Brow = (const v2f*)(cent + rB * DIM);

    v8f cacc = {};
#pragma unroll
    for (int kc = 0; kc < 8; ++kc) {
      v2f a = Arow[kc * 2 + h];           // K = {4kc+2h, 4kc+2h+1}
      v2f b = Brow[kc * 2 + h];
      a *= maskA;
      b *= maskB;
      cacc = __builtin_amdgcn_wmma_f32_16x16x4_f32(
          /*neg_a=*/false, a, /*neg_b=*/false, b,
          /*c_mod=*/(short)0, cacc, /*reuse_a=*/false, /*reuse_b=*/false);
    }

    const int j  = tj * 16 + m;           // column index for this lane
    const int jc = (j < CLS) ? j : (CLS - 1);
    const float sqj = sq[jc];
    const float svj = sv[jc];
#pragma unroll
    for (int v = 0; v < 8; ++v) {
      int i  = ti * 16 + v + 8 * h;       // C/D layout: VGPR v, lane-half h
      int ic = (i < CLS) ? i : (CLS - 1);
      float gg = cacc[v];
      float d2 = sq[ic] + sqj - 2.0f * gg;
      float mm = sqrtf(fmaxf(d2, 0.0f));
      float r  = (sv[ic] + svj) / mm;
      bool valid = (i < CLS) && (j < CLS) && (i != j);
      acc += valid ? r : 0.0f;
    }
  }

  // fixed-structure wave reduction (deterministic)
#pragma unroll
  for (int off = 16; off > 0; off >>= 1) acc += __shfl_xor(acc, off);
  if (lane == 0) wsum[waveId] = acc;
  __syncthreads();
  if (threadIdx.x == 0) {
    float s = 0.0f;
#pragma unroll
    for (int w = 0; w < WAVES_PER_BLOCK; ++w) s += wsum[w];
    ws[WS_BLK + blockIdx.x] = s;
  }
}

// ---------------------------------------------------------------------------
// K6: deterministic tree reduction of per-block sums; out = total / C.
// ---------------------------------------------------------------------------
__global__ void k_reduce(const float* __restrict__ ws, float* __restrict__ out,
                         int nblk) {
  __shared__ float sm[512];
  int tid = threadIdx.x;
  sm[tid] = (tid < nblk) ? ws[WS_BLK + tid] : 0.0f;
  __syncthreads();
  for (int s = 256; s > 0; s >>= 1) {
    if (tid < s) sm[tid] += sm[tid + s];
    __syncthreads();
  }
  if (tid == 0) out[0] = sm[0] / (float)CLS;
}

// ---------------------------------------------------------------------------
extern "C" void kernel_launch(void* const* d_in, const int* in_sizes, int n_in,
                              void* d_out, int out_size, void* d_ws, size_t ws_size,
                              hipStream_t stream) {
  const float* predicted = (const float*)d_in[0]; // N x 32
  const float* centroids = (const float*)d_in[1]; // C x 32
  const float* distances = (const float*)d_in[2]; // C x 1
  const float* count     = (const float*)d_in[3]; // C x 1
  const int*   target    = (const int*)d_in[4];   // N
  float* ws  = (float*)d_ws;
  float* out = (float*)d_out;

  (void)in_sizes; (void)n_in; (void)out_size; (void)ws_size;

  // K1: init accumulators (32000 = 125 * 256 exactly)
  k_init<<<125, 256, 0, stream>>>(centroids, distances, ws);

  // K2: scatter-add contribs into cent2 (LDS-staged). 128 KB dynamic LDS.
  k_scatter_cent<<<256, 256, CLS * DIM * sizeof(float), stream>>>(
      predicted, count, target, ws);

  // K3: scatter-add per-point norms into s (LDS-staged). 4 KB dynamic LDS.
  k_scatter_norm<<<256, 256, CLS * sizeof(float), stream>>>(
      predicted, count, target, ws);

  // K4: sv / sq
  k_svsq<<<(CLS + 255) / 256, 256, 0, stream>>>(count, ws);

  // K5: WMMA pairwise tile kernel
  const int nblkPair = (TILE_PAIRS + WAVES_PER_BLOCK - 1) / WAVES_PER_BLOCK; // 497
  k_pair<<<nblkPair, 256, 0, stream>>>(ws);

  // K6: final deterministic reduction
  k_reduce<<<1, 512, 0, stream>>>(ws, out, nblkPair);
}